// XFeatBEVNet_33371895890580
// MI455X (gfx1250) — compile-verified
//
#include <hip/hip_runtime.h>
#include <hip/hip_bf16.h>

// ---------------- LSS grid constants (match reference) ----------------
#define FW 44
#define FH 16
#define ND 41
#define NN 6
#define NB 4
#define NC 64
#define NX0 200
#define NX1 200
#define PTS_PER_BLOCK 128
#define NPRIME (NB * NN * ND * FH * FW) // 692736
// floats per camera param block: Rpinv[9], M[9], trans[3], post_trans[3]
#define PBLK 24

__device__ __forceinline__ void inv3x3(const float* __restrict__ m, float* __restrict__ o) {
    const float a = m[0], b = m[1], c = m[2];
    const float d = m[3], e = m[4], f = m[5];
    const float g = m[6], h = m[7], i = m[8];
    const float A = e * i - f * h;
    const float B = -(d * i - f * g);
    const float C = d * h - e * g;
    const float det = a * A + b * B + c * C;
    const float r = 1.0f / det;
    o[0] = A * r;
    o[1] = -(b * i - c * h) * r;
    o[2] = (b * f - c * e) * r;
    o[3] = B * r;
    o[4] = (a * i - c * g) * r;
    o[5] = -(a * f - c * d) * r;
    o[6] = C * r;
    o[7] = -(a * h - b * g) * r;
    o[8] = (a * e - b * d) * r;
}

// Precompute per-(b,n) fused camera params into workspace:
//   Rpinv = inv(post_rots), M = rots @ inv(intrins), trans, post_trans
__global__ void lss_setup_kernel(const float* __restrict__ rots,
                                 const float* __restrict__ trans,
                                 const float* __restrict__ intrins,
                                 const float* __restrict__ post_rots,
                                 const float* __restrict__ post_trans,
                                 float* __restrict__ P) {
    const int k = threadIdx.x;
    if (k >= NB * NN) return;
    float Kinv[9], Rpinv[9], M[9];
    inv3x3(intrins + k * 9, Kinv);
    inv3x3(post_rots + k * 9, Rpinv);
    const float* R = rots + k * 9;
#pragma unroll
    for (int r = 0; r < 3; ++r)
#pragma unroll
        for (int c = 0; c < 3; ++c)
            M[r * 3 + c] = R[r * 3 + 0] * Kinv[0 + c] +
                           R[r * 3 + 1] * Kinv[3 + c] +
                           R[r * 3 + 2] * Kinv[6 + c];
    float* o = P + k * PBLK;
#pragma unroll
    for (int j = 0; j < 9; ++j) o[j] = Rpinv[j];
#pragma unroll
    for (int j = 0; j < 9; ++j) o[9 + j] = M[j];
#pragma unroll
    for (int j = 0; j < 3; ++j) o[18 + j] = trans[k * 3 + j];
#pragma unroll
    for (int j = 0; j < 3; ++j) o[21 + j] = post_trans[k * 3 + j];
}

// Main kernel: async-stage a 128-point x-tile (32 KB) into LDS, overlap with
// geometry + voxel-index computation, then scatter-add 64 channels/point with
// global f32 atomics (output fits in the 192 MB L2).
__global__ __launch_bounds__(256) void lss_scatter_kernel(const float* __restrict__ x,
                                                          const float* __restrict__ P,
                                                          float* __restrict__ out) {
    __shared__ float sfeat[PTS_PER_BLOCK * NC]; // 32 KB feature staging
    __shared__ int sidx[PTS_PER_BLOCK];         // per-point output base (or -1)

    const int tid = threadIdx.x;
    const int pbase = blockIdx.x * PTS_PER_BLOCK;

    // ---- 1) issue async DMA: global -> LDS (tracked by ASYNCcnt) ----
    {
        const unsigned lds_base = (unsigned)(size_t)(void*)sfeat;
        const float4* src = (const float4*)x + (size_t)pbase * (NC / 4);
        for (int i = tid; i < PTS_PER_BLOCK * NC / 4; i += 256) {
            const unsigned laddr = lds_base + (unsigned)i * 16u;
            const float4* ga = src + i;
            asm volatile("global_load_async_to_lds_b128 %0, %1, off"
                         :: "v"(laddr), "v"(ga)
                         : "memory");
        }
    }

    // ---- 2) geometry + voxelization, overlapped with the DMA ----
    if (tid < PTS_PER_BLOCK) {
        int pidx = pbase + tid;
        const int w = pidx % FW;
        int t = pidx / FW;
        const int h = t % FH; t /= FH;
        const int d = t % ND; t /= ND;
        const int n = t % NN;
        const int b = t / NN;
        const float* pp = P + (b * NN + n) * PBLK;

        const float xs = (float)w * (703.0f / 43.0f); // linspace(0,703,44)
        const float ys = (float)h * 17.0f;            // linspace(0,255,16)
        const float zs = 4.0f + (float)d;             // arange(4,45,1)

        const float px = xs - pp[21];
        const float py = ys - pp[22];
        const float pz = zs - pp[23];
        // q = post_rots_inv @ p
        const float qx = fmaf(pp[0], px, fmaf(pp[1], py, pp[2] * pz));
        const float qy = fmaf(pp[3], px, fmaf(pp[4], py, pp[5] * pz));
        const float qz = fmaf(pp[6], px, fmaf(pp[7], py, pp[8] * pz));
        // perspective: (x*z, y*z, z)
        const float ux = qx * qz, uy = qy * qz, uz = qz;
        // geom = (rots @ intrins_inv) @ u + trans
        const float gx = fmaf(pp[9], ux, fmaf(pp[10], uy, fmaf(pp[11], uz, pp[18])));
        const float gy = fmaf(pp[12], ux, fmaf(pp[13], uy, fmaf(pp[14], uz, pp[19])));
        const float gz = fmaf(pp[15], ux, fmaf(pp[16], uy, fmaf(pp[17], uz, pp[20])));

        // voxelize: (g - (BX - DX/2)) / DX ; clip +-1e6 ; trunc toward zero
        float fx = (gx + 50.0f) * 2.0f;
        float fy = (gy + 50.0f) * 2.0f;
        float fz = (gz + 10.0f) * (1.0f / 20.0f);
        fx = fminf(fmaxf(fx, -1.0e6f), 1.0e6f);
        fy = fminf(fmaxf(fy, -1.0e6f), 1.0e6f);
        fz = fminf(fmaxf(fz, -1.0e6f), 1.0e6f);
        const int vx = (int)truncf(fx);
        const int vy = (int)truncf(fy);
        const int vz = (int)truncf(fz);
        const bool kept = (vx >= 0) && (vx < NX0) && (vy >= 0) && (vy < NX1) &&
                          (vz >= 0) && (vz < 1);
        // out[b][c][vx][vy]  ->  base = b*NC*40000 + vx*200 + vy  (+ c*40000 later)
        sidx[tid] = kept ? (b * NC * (NX0 * NX1) + vx * NX1 + vy) : -1;
    }

    // ---- 3) wait this wave's async DMA, then block barrier ----
    asm volatile("s_wait_asynccnt 0" ::: "memory");
    __syncthreads();

    // ---- 4) scatter: 2 threads per point, 32 channels each ----
    const int p = tid >> 1;
    const int c0 = (tid & 1) * 32;
    const int base = sidx[p];
    if (base >= 0) {
#pragma unroll 4
        for (int j = 0; j < 32; ++j) {
            // additive channel swizzle -> conflict-free across 64 LDS banks
            const int c = (c0 + j + p) & (NC - 1);
            atomicAdd(out + base + c * (NX0 * NX1), sfeat[p * NC + c]);
        }
    }
}

extern "C" void kernel_launch(void* const* d_in, const int* in_sizes, int n_in,
                              void* d_out, int out_size, void* d_ws, size_t ws_size,
                              hipStream_t stream) {
    (void)in_sizes; (void)n_in; (void)ws_size;
    const float* x          = (const float*)d_in[0];
    const float* rots       = (const float*)d_in[1];
    const float* trans      = (const float*)d_in[2];
    const float* intrins    = (const float*)d_in[3];
    const float* post_rots  = (const float*)d_in[4];
    const float* post_trans = (const float*)d_in[5];
    float* out = (float*)d_out;
    float* P = (float*)d_ws;

    hipMemsetAsync(out, 0, (size_t)out_size * sizeof(float), stream);
    lss_setup_kernel<<<1, 32, 0, stream>>>(rots, trans, intrins, post_rots, post_trans, P);
    lss_scatter_kernel<<<NPRIME / PTS_PER_BLOCK, 256, 0, stream>>>(x, P, out);
}